// MoEBiasLayer_46883863003306
// MI455X (gfx1250) — compile-verified
//
#include <hip/hip_runtime.h>
#include <hip/hip_bf16.h>
#include <math.h>

typedef __attribute__((ext_vector_type(2))) float v2f;
typedef __attribute__((ext_vector_type(4))) float v4f;
typedef __attribute__((ext_vector_type(8))) float v8f;

#define NUM_EXPERTS 16
#define T_TOKENS    2048
#define H_DIM       2048
#define V_DIM       32000

// ---------------------------------------------------------------------------
// Kernel 0: zero the per-expert usage accumulators in workspace
// ---------------------------------------------------------------------------
__global__ void moe_init_kernel(float* __restrict__ expert_sum) {
    if (threadIdx.x < NUM_EXPERTS) expert_sum[threadIdx.x] = 0.0f;
}

// ---------------------------------------------------------------------------
// Kernel 1: gating.  One block = 16 tokens.  4 waves split K=2048 into
// quarters; each wave accumulates the full 16x16 (token x expert) logits tile
// with V_WMMA_F32_16X16X4_F32, partial tiles reduced through LDS.
// Then softmax / top-2 / renormalize per token, store weights+indices to ws,
// and accumulate per-expert score sums (for aux loss) with one atomic per
// expert per block.
// ---------------------------------------------------------------------------
__global__ void __launch_bounds__(128)
moe_gate_kernel(const float* __restrict__ hs,      // (T, H)
                const float* __restrict__ gw,      // (E, H)
                float* __restrict__ topk_w,        // (T, 2)
                int*   __restrict__ topk_i,        // (T, 2)
                float* __restrict__ expert_sum) {  // (E,)
    __shared__ float partial[4][32][8];            // wave x lane x acc-vgpr
    __shared__ float logits[16][NUM_EXPERTS + 1];  // token x expert (padded)
    __shared__ float scores[16][NUM_EXPERTS + 1];  // token x expert (padded)

    const int lane   = threadIdx.x & 31;
    const int wave   = threadIdx.x >> 5;
    const int grp    = lane >> 4;          // half-wave group (0 or 1)
    const int rowcol = lane & 15;          // A row (token) / B col (expert)
    const int koff   = grp << 1;           // K offset within 4-wide step: 0 or 2
    const int tok0   = blockIdx.x * 16;

    const float* aRow = hs + (size_t)(tok0 + rowcol) * H_DIM;  // token row
    const float* bRow = gw + (size_t)rowcol * H_DIM;           // expert row

    // Each wave covers a contiguous quarter of K.
    const int kBeg = wave * (H_DIM / 4);
    const int kEnd = kBeg + (H_DIM / 4);

    v8f c = {};
    #pragma unroll 4
    for (int k = kBeg; k < kEnd; k += 4) {
        // 32-bit A 16x4 layout: lanes 0-15 hold K={k,k+1}, lanes 16-31 K={k+2,k+3}
        v2f a = *(const v2f*)(aRow + k + koff);
        v2f b = *(const v2f*)(bRow + k + koff);
        c = __builtin_amdgcn_wmma_f32_16x16x4_f32(
                /*neg_a=*/false, a, /*neg_b=*/false, b,
                /*c_mod=*/(short)0, c, /*reuse_a=*/false, /*reuse_b=*/false);
    }

    #pragma unroll
    for (int v = 0; v < 8; ++v) partial[wave][lane][v] = c[v];
    __syncthreads();

    // Wave 0 reduces the 4 partial tiles and scatters into [token][expert] form.
    if (wave == 0) {
        #pragma unroll
        for (int v = 0; v < 8; ++v) {
            float s = partial[0][lane][v] + partial[1][lane][v] +
                      partial[2][lane][v] + partial[3][lane][v];
            // C/D layout: lanes 0-15 -> M = v, lanes 16-31 -> M = v + 8; N = lane&15
            logits[v + (grp << 3)][rowcol] = s;
        }
    }
    __syncthreads();

    // One lane per token: softmax over 16 experts, top-2, renormalize.
    if (wave == 0 && lane < 16) {
        float l[NUM_EXPERTS];
        float m = -3.4e38f;
        #pragma unroll
        for (int e = 0; e < NUM_EXPERTS; ++e) {
            l[e] = logits[lane][e];
            m = fmaxf(m, l[e]);
        }
        float s = 0.0f;
        #pragma unroll
        for (int e = 0; e < NUM_EXPERTS; ++e) { l[e] = expf(l[e] - m); s += l[e]; }
        const float inv = 1.0f / s;

        int   i0 = 0; float s0 = -1.0f;
        #pragma unroll
        for (int e = 0; e < NUM_EXPERTS; ++e) {
            float sc = l[e] * inv;
            scores[lane][e] = sc;
            if (sc > s0) { s0 = sc; i0 = e; }
        }
        int   i1 = 0; float s1 = -1.0f;
        #pragma unroll
        for (int e = 0; e < NUM_EXPERTS; ++e) {
            float sc = scores[lane][e];
            if (e != i0 && sc > s1) { s1 = sc; i1 = e; }
        }
        const float wn = 1.0f / (s0 + s1);
        const int t = tok0 + lane;
        topk_w[2 * t + 0] = s0 * wn;
        topk_w[2 * t + 1] = s1 * wn;
        topk_i[2 * t + 0] = i0;
        topk_i[2 * t + 1] = i1;
    }
    __syncthreads();

    // Lane = expert: sum scores over this block's 16 tokens, one atomic/expert.
    if (wave == 0 && lane < NUM_EXPERTS) {
        float acc = 0.0f;
        #pragma unroll
        for (int t = 0; t < 16; ++t) acc += scores[t][lane];
        atomicAdd(&expert_sum[lane], acc);
    }
}

// ---------------------------------------------------------------------------
// Kernel 2: streaming bias combine.  One block per token; float4 loads of the
// two selected expert rows (L2-resident: expert_biases is only 2 MB) and
// non-temporal float4 stores of the 262 MB output so it doesn't evict them.
// ---------------------------------------------------------------------------
__global__ void __launch_bounds__(256)
moe_bias_kernel(const float* __restrict__ eb,      // (E, V)
                const float* __restrict__ topk_w,  // (T, 2)
                const int*   __restrict__ topk_i,  // (T, 2)
                float* __restrict__ out) {         // (T, V)
    const int t  = blockIdx.x;
    const float w0 = topk_w[2 * t + 0];
    const float w1 = topk_w[2 * t + 1];
    const int   i0 = topk_i[2 * t + 0];
    const int   i1 = topk_i[2 * t + 1];

    const v4f* __restrict__ b0 = (const v4f*)(eb + (size_t)i0 * V_DIM);
    const v4f* __restrict__ b1 = (const v4f*)(eb + (size_t)i1 * V_DIM);
    v4f* __restrict__ o = (v4f*)(out + (size_t)t * V_DIM);

    for (int v = threadIdx.x; v < V_DIM / 4; v += 256) {
        v4f x = b0[v];
        v4f y = b1[v];
        v4f r = w0 * x + w1 * y;
        __builtin_nontemporal_store(r, &o[v]);
    }
}

// ---------------------------------------------------------------------------
// Kernel 3: aux loss scalar from the accumulated expert usage sums.
// ---------------------------------------------------------------------------
__global__ void moe_aux_kernel(const float* __restrict__ expert_sum,
                               float* __restrict__ out_aux) {
    if (threadIdx.x == 0) {
        float aux = 0.0f;
        #pragma unroll
        for (int e = 0; e < NUM_EXPERTS; ++e) {
            float u = expert_sum[e] * (1.0f / (float)T_TOKENS);
            aux += u * logf(u);
        }
        *out_aux = aux * (float)NUM_EXPERTS;
    }
}

// ---------------------------------------------------------------------------
extern "C" void kernel_launch(void* const* d_in, const int* in_sizes, int n_in,
                              void* d_out, int out_size, void* d_ws, size_t ws_size,
                              hipStream_t stream) {
    const float* hs = (const float*)d_in[0];   // (T, H)  fp32
    const float* gw = (const float*)d_in[1];   // (E, H)  fp32
    const float* eb = (const float*)d_in[2];   // (E, V)  fp32
    float* out = (float*)d_out;                // bias (T,V) then aux scalar

    // Workspace layout: weights (2T f32) | indices (2T i32) | expert sums (16 f32)
    float* topk_w     = (float*)d_ws;
    int*   topk_i     = (int*)(topk_w + 2 * T_TOKENS);
    float* expert_sum = (float*)(topk_i + 2 * T_TOKENS);

    moe_init_kernel<<<1, 32, 0, stream>>>(expert_sum);
    moe_gate_kernel<<<T_TOKENS / 16, 128, 0, stream>>>(hs, gw, topk_w, topk_i, expert_sum);
    moe_bias_kernel<<<T_TOKENS, 256, 0, stream>>>(eb, topk_w, topk_i, out);
    moe_aux_kernel<<<1, 32, 0, stream>>>(expert_sum, out + (size_t)T_TOKENS * V_DIM);
}